// GCN_81286551044217
// MI455X (gfx1250) — compile-verified
//
#include <hip/hip_runtime.h>
#include <hip/hip_bf16.h>

typedef __attribute__((ext_vector_type(2))) float v2f;
typedef __attribute__((ext_vector_type(8))) float v8f;

// ---------------------------------------------------------------------------
// Utility kernels
// ---------------------------------------------------------------------------
__global__ void zero_i32_kernel(int* __restrict__ p, int n) {
    int i = blockIdx.x * blockDim.x + threadIdx.x;
    if (i < n) p[i] = 0;
}

// In-degree at dst over real edges (self-loops folded in later as +1).
__global__ void degree_kernel(const long long* __restrict__ dsts, int E,
                              int* __restrict__ cnt) {
    int e = blockIdx.x * blockDim.x + threadIdx.x;
    if (e < E) atomicAdd(&cnt[(int)dsts[e]], 1);
}

// dinv[i] = rsqrt(deg[i]) with deg = cnt + 1 (self-loop) >= 1 always.
// In-place: reads the int count, writes the float (each thread owns one slot).
__global__ void dinv_kernel(const int* __restrict__ cnt, float* __restrict__ dinv,
                            int n) {
    int i = blockIdx.x * blockDim.x + threadIdx.x;
    if (i < n) {
        int c = cnt[i];
        dinv[i] = rsqrtf((float)(c + 1));
    }
}

// agg[i][f] = b[f] + h[i][f] * dinv[i]^2   (bias + self-loop message)
template <int F>
__global__ void init_agg_kernel(const float* __restrict__ h,
                                const float* __restrict__ dinv,
                                const float* __restrict__ bias,
                                float* __restrict__ agg, int n) {
    int idx = blockIdx.x * blockDim.x + threadIdx.x;
    if (idx < n * F) {
        int i = idx / F;
        int f = idx - i * F;
        float di = dinv[i];
        agg[idx] = bias[f] + h[idx] * di * di;
    }
}

// Edge scatter: wave-cooperative, F lanes per edge -> one coalesced line of
// h[src] loaded and one coalesced line of global_atomic_add_f32 into agg[dst].
template <int F>
__global__ void scatter_kernel(const float* __restrict__ h,
                               const long long* __restrict__ srcs,
                               const long long* __restrict__ dsts,
                               const float* __restrict__ dinv,
                               float* __restrict__ agg, int E) {
    int t    = blockIdx.x * blockDim.x + threadIdx.x;
    int lane = t & 31;
    int w    = t >> 5;
    constexpr int EPW = 32 / F;           // edges per wave
    int e = w * EPW + (lane / F);
    int f = lane & (F - 1);
    if (e < E) {
        int s = (int)srcs[e];
        int d = (int)dsts[e];
        float val = h[(size_t)s * F + f] * dinv[s] * dinv[d];
        atomicAdd(&agg[(size_t)d * F + f], val);
    }
}

// ---------------------------------------------------------------------------
// Exact-fp32 GEMM via V_WMMA_F32_16X16X4_F32.
// H[nrows x NOUT] = act(X[nrows x K]) @ W[K x NOUT]
// One wave per 16-row tile; NOUT <= 32 handled with 1-2 accumulators
// (columns >= NOUT padded with zero B columns).
// A frag: lane m = lane&15 -> row m; vgpr j holds K = k + 2*(lane>>4) + j.
// B frag: lane -> column (m);        vgpr j holds K = k + 2*(lane>>4) + j.
// D frag: vgpr r -> row r + 8*(lane>>4), column m.
// ---------------------------------------------------------------------------
template <int K, int NOUT, bool RELUIN>
__global__ void gemm_wmma_kernel(const float* __restrict__ X,
                                 const float* __restrict__ W,
                                 float* __restrict__ H, int nrows) {
    int wave = blockIdx.x * (blockDim.x >> 5) + (threadIdx.x >> 5);
    int ntiles = (nrows + 15) >> 4;
    if (wave >= ntiles) return;

    int lane = threadIdx.x & 31;
    int half = lane >> 4;     // 0: K pair {0,1}; 1: K pair {2,3}
    int m    = lane & 15;

    int arow = wave * 16 + m;
    if (arow >= nrows) arow = nrows - 1;          // safe clamp (stores guarded)
    const float* xr = X + (size_t)arow * K;

    v8f acc0 = {};
    v8f acc1 = {};

#pragma unroll
    for (int k = 0; k < K; k += 4) {
        int kk = k + 2 * half;
        float a0 = xr[kk];
        float a1 = xr[kk + 1];
        if (RELUIN) { a0 = fmaxf(a0, 0.0f); a1 = fmaxf(a1, 0.0f); }
        v2f a = {a0, a1};

        float b00 = (m < NOUT) ? W[(size_t)kk * NOUT + m]       : 0.0f;
        float b01 = (m < NOUT) ? W[(size_t)(kk + 1) * NOUT + m] : 0.0f;
        v2f b0 = {b00, b01};
        acc0 = __builtin_amdgcn_wmma_f32_16x16x4_f32(
            false, a, false, b0, (short)0, acc0, false, false);

        if (NOUT > 16) {
            v2f b1 = {W[(size_t)kk * NOUT + 16 + m],
                      W[(size_t)(kk + 1) * NOUT + 16 + m]};
            acc1 = __builtin_amdgcn_wmma_f32_16x16x4_f32(
                false, a, false, b1, (short)0, acc1, false, false);
        }
    }

    int baserow = wave * 16 + 8 * half;
#pragma unroll
    for (int r = 0; r < 8; r++) {
        int row = baserow + r;
        if (row < nrows) {
            if (m < NOUT) H[(size_t)row * NOUT + m] = acc0[r];
            if (NOUT > 16) H[(size_t)row * NOUT + 16 + m] = acc1[r];
        }
    }
}

// ---------------------------------------------------------------------------
// Launch
// ---------------------------------------------------------------------------
static inline int cdiv(int a, int b) { return (a + b - 1) / b; }

extern "C" void kernel_launch(void* const* d_in, const int* in_sizes, int n_in,
                              void* d_out, int out_size, void* d_ws, size_t ws_size,
                              hipStream_t stream) {
    const float*     x    = (const float*)d_in[0];
    const long long* ei   = (const long long*)d_in[1];   // int64 [2, E]
    const float*     W1   = (const float*)d_in[2];
    const float*     b1   = (const float*)d_in[3];
    const float*     W2   = (const float*)d_in[4];
    const float*     b2   = (const float*)d_in[5];
    const float*     W3   = (const float*)d_in[6];
    const float*     b3   = (const float*)d_in[7];

    const int N = in_sizes[0] / 128;
    const int E = in_sizes[1] / 2;
    const long long* srcs = ei;
    const long long* dsts = ei + E;

    // Workspace layout
    char*  ws   = (char*)d_ws;
    size_t o0   = 0;                                   // cnt / dinv : N * 4
    size_t o1   = ((size_t)N * 4 + 255) & ~(size_t)255; // h : N * 32 * 4
    size_t o2   = o1 + ((size_t)N * 32 * 4 + 255 & ~(size_t)255); // agg : N*32*4
    int*   cnt  = (int*)(ws + o0);
    float* dinv = (float*)(ws + o0);
    float* h    = (float*)(ws + o1);
    float* agg  = (float*)(ws + o2);
    float* out  = (float*)d_out;

    const int B = 256;

    // --- normalization ---
    zero_i32_kernel<<<cdiv(N, B), B, 0, stream>>>(cnt, N);
    degree_kernel<<<cdiv(E, B), B, 0, stream>>>(dsts, E, cnt);
    dinv_kernel<<<cdiv(N, B), B, 0, stream>>>(cnt, dinv, N);

    const int ntiles = cdiv(N, 16);
    const int gblocks = cdiv(ntiles, B / 32);

    // --- layer 1: 128 -> 32 ---
    gemm_wmma_kernel<128, 32, false><<<gblocks, B, 0, stream>>>(x, W1, h, N);
    init_agg_kernel<32><<<cdiv(N * 32, B), B, 0, stream>>>(h, dinv, b1, agg, N);
    {
        int waves = cdiv(E, 1);                       // 1 edge per wave (F=32)
        scatter_kernel<32><<<cdiv(waves * 32, B), B, 0, stream>>>(
            h, srcs, dsts, dinv, agg, E);
    }

    // --- layer 2: 32 -> 16 (ReLU on input, fused) ---
    gemm_wmma_kernel<32, 16, true><<<gblocks, B, 0, stream>>>(agg, W2, h, N);
    init_agg_kernel<16><<<cdiv(N * 16, B), B, 0, stream>>>(h, dinv, b2, agg, N);
    {
        int waves = cdiv(E, 2);                       // 2 edges per wave (F=16)
        scatter_kernel<16><<<cdiv(waves * 32, B), B, 0, stream>>>(
            h, srcs, dsts, dinv, agg, E);
    }

    // --- layer 3: 16 -> 8 (ReLU on input, fused), agg directly into d_out ---
    gemm_wmma_kernel<16, 8, true><<<gblocks, B, 0, stream>>>(agg, W3, h, N);
    init_agg_kernel<8><<<cdiv(N * 8, B), B, 0, stream>>>(h, dinv, b3, out, N);
    {
        int waves = cdiv(E, 4);                       // 4 edges per wave (F=8)
        scatter_kernel<8><<<cdiv(waves * 32, B), B, 0, stream>>>(
            h, srcs, dsts, dinv, out, E);
    }
}